// GNNLayer_27230092657474
// MI455X (gfx1250) — compile-verified
//
#include <hip/hip_runtime.h>

#define U_CNT 100000
#define I_CNT 50000
#define DIM   128
#define B_CNT 4
#define E_CNT 500000

typedef __attribute__((ext_vector_type(2))) float v2f;
typedef __attribute__((ext_vector_type(8))) float v8f;

// ---------------------------------------------------------------------------
// Zero the fp32 accumulators in workspace (poisoned 0xAA by harness).
// ---------------------------------------------------------------------------
__global__ void zero_kernel(float* __restrict__ p, long n /* multiple of 4 */) {
    long n4 = n >> 2;
    float4* p4 = (float4*)p;
    long stride = (long)gridDim.x * blockDim.x;
    for (long k = (long)blockIdx.x * blockDim.x + threadIdx.x; k < n4; k += stride)
        p4[k] = make_float4(0.f, 0.f, 0.f, 0.f);
}

// ---------------------------------------------------------------------------
// Edge scatter: one wave32 per edge, each lane owns 4 consecutive dwords of
// the 128-float embedding. Hardware fp32 atomics (global_atomic_add_f32).
//   uacc[b][u][:] += w * item_emb[i][:]
//   iacc[b][i][:] += w * user_emb[u][:]
// ---------------------------------------------------------------------------
__global__ __launch_bounds__(256)
void scatter_kernel(const float* __restrict__ user_emb,
                    const float* __restrict__ item_emb,
                    const int*   __restrict__ eu,
                    const int*   __restrict__ ei,
                    const float* __restrict__ ew,
                    float* __restrict__ uacc,
                    float* __restrict__ iacc) {
    long gid  = (long)blockIdx.x * blockDim.x + threadIdx.x;
    long e    = gid >> 5;                 // edge id (wave32 per edge)
    int  lane = threadIdx.x & 31;
    if (e >= (long)B_CNT * E_CNT) return;

    int   u  = eu[e];
    int   it = ei[e];
    float w  = ew[e];
    long  b  = e / E_CNT;

    const float4 vi = ((const float4*)(item_emb + (long)it * DIM))[lane];
    const float4 vu = ((const float4*)(user_emb + (long)u  * DIM))[lane];

    float* du = uacc + ((b * U_CNT + u ) * (long)DIM) + lane * 4;
    float* di = iacc + ((b * I_CNT + it) * (long)DIM) + lane * 4;

    unsafeAtomicAdd(du + 0, w * vi.x);
    unsafeAtomicAdd(du + 1, w * vi.y);
    unsafeAtomicAdd(du + 2, w * vi.z);
    unsafeAtomicAdd(du + 3, w * vi.w);

    unsafeAtomicAdd(di + 0, w * vu.x);
    unsafeAtomicAdd(di + 1, w * vu.y);
    unsafeAtomicAdd(di + 2, w * vu.z);
    unsafeAtomicAdd(di + 3, w * vu.w);
}

// ---------------------------------------------------------------------------
// Fused GEMM + PReLU using V_WMMA_F32_16X16X4_F32 (wave32).
// Per workgroup (8 waves, one 16-row tile):
//   - stage weight W[128][128] in LDS (padded rows)
//   - stage 4 behavior A-tiles [16][128] + build the mean tile in LDS
//   - wave w computes column panel n0=w*16 for all 5 A matrices:
//       32 x wmma_f32_16x16x4_f32 (K=128), PReLU, store.
// ---------------------------------------------------------------------------
__global__ __launch_bounds__(256)
void gemm_prelu_kernel(const float* __restrict__ acc,      // [B][R][D]
                       const float* __restrict__ W,        // [D][D]
                       const float* __restrict__ alpha_p,  // [1]
                       float* __restrict__ multi_out,      // [R][D]
                       float* __restrict__ single_out,     // [B][R][D]
                       int R) {
    constexpr int LDW = DIM + 16;  // 144: splits half-wave bank groups
    __shared__ float sW[DIM * LDW];              // 73,728 B
    __shared__ float sA[(B_CNT + 1) * 16 * LDW]; // 46,080 B

    const int   tid   = threadIdx.x;
    const long  row0  = (long)blockIdx.x * 16;
    const float alpha = alpha_p[0];

    // ---- stage weight (float4 vectorized) ----
    for (int idx = tid; idx < DIM * (DIM / 4); idx += 256) {
        int k  = idx >> 5;
        int n4 = (idx & 31) << 2;
        float4 v = *(const float4*)(W + (long)k * DIM + n4);
        *(float4*)(&sW[k * LDW + n4]) = v;
    }
    // ---- stage the 4 behavior A-tiles ----
    for (int b = 0; b < B_CNT; ++b) {
        for (int idx = tid; idx < 16 * (DIM / 4); idx += 256) {
            int m  = idx >> 5;
            int k4 = (idx & 31) << 2;
            float4 v = *(const float4*)(acc + (((long)b * R + row0 + m) * DIM) + k4);
            *(float4*)(&sA[(b * 16 + m) * LDW + k4]) = v;
        }
    }
    __syncthreads();
    // ---- mean tile (behavior slot B_CNT) ----
    for (int idx = tid; idx < 16 * DIM; idx += 256) {
        int m = idx >> 7;
        int k = idx & (DIM - 1);
        float s = sA[(0 * 16 + m) * LDW + k] + sA[(1 * 16 + m) * LDW + k] +
                  sA[(2 * 16 + m) * LDW + k] + sA[(3 * 16 + m) * LDW + k];
        sA[(B_CNT * 16 + m) * LDW + k] = 0.25f * s;
    }
    __syncthreads();

    const int wave  = tid >> 5;
    const int lane  = tid & 31;
    const int n0    = wave * 16;
    const int ncol  = n0 + (lane & 15);
    const int khalf = (lane >> 4) * 2;  // 0 for lanes 0-15, 2 for lanes 16-31
    const int mtop  = (lane >> 4) * 8;  // C/D: VGPR r holds rows r and r+8
    const int arow  = lane & 15;

    for (int am = 0; am <= B_CNT; ++am) {
        const float* A = &sA[am * 16 * LDW];
        v8f c = {};
        #pragma unroll
        for (int ks = 0; ks < DIM / 4; ++ks) {
            const int k0 = ks * 4 + khalf;
            v2f a;
            a.x = A[arow * LDW + k0];
            a.y = A[arow * LDW + k0 + 1];
            v2f bf;
            bf.x = sW[(k0)     * LDW + ncol];
            bf.y = sW[(k0 + 1) * LDW + ncol];
            // D = A(16x4,f32) * B(4x16,f32) + C(16x16,f32)
            c = __builtin_amdgcn_wmma_f32_16x16x4_f32(
                    /*neg_a=*/false, a, /*neg_b=*/false, bf,
                    /*c_mod=*/(short)0, c, /*reuse_a=*/false, /*reuse_b=*/false);
        }
        float* dst = (am == B_CNT)
                         ? (multi_out + row0 * DIM)
                         : (single_out + (((long)am * R + row0) * DIM));
        #pragma unroll
        for (int r = 0; r < 8; ++r) {
            float v = c[r];
            v = (v >= 0.f) ? v : alpha * v;
            dst[(long)(r + mtop) * DIM + ncol] = v;
        }
    }
}

// ---------------------------------------------------------------------------
extern "C" void kernel_launch(void* const* d_in, const int* in_sizes, int n_in,
                              void* d_out, int out_size, void* d_ws, size_t ws_size,
                              hipStream_t stream) {
    const float* user_emb = (const float*)d_in[0];  // [U,D]
    const float* item_emb = (const float*)d_in[1];  // [I,D]
    const int*   eu       = (const int*)  d_in[2];  // [B,E]
    const int*   ei       = (const int*)  d_in[3];  // [B,E]
    const float* ew       = (const float*)d_in[4];  // [B,E]
    const float* u_w      = (const float*)d_in[5];  // [D,D]
    const float* i_w      = (const float*)d_in[6];  // [D,D]
    const float* alpha    = (const float*)d_in[7];  // [1]

    float* uacc = (float*)d_ws;                               // [B,U,D]
    float* iacc = uacc + (long)B_CNT * U_CNT * DIM;           // [B,I,D]

    float* multi_u  = (float*)d_out;                          // [U,D]
    float* multi_i  = multi_u  + (long)U_CNT * DIM;           // [I,D]
    float* single_u = multi_i  + (long)I_CNT * DIM;           // [B,U,D]
    float* single_i = single_u + (long)B_CNT * U_CNT * DIM;   // [B,I,D]

    const long accN = (long)B_CNT * (U_CNT + I_CNT) * DIM;
    zero_kernel<<<2048, 256, 0, stream>>>(uacc, accN);

    const long totalEdges = (long)B_CNT * E_CNT;
    const long threads    = totalEdges * 32;
    const int  blocks     = (int)((threads + 255) / 256);
    scatter_kernel<<<blocks, 256, 0, stream>>>(user_emb, item_emb, eu, ei, ew,
                                               uacc, iacc);

    gemm_prelu_kernel<<<U_CNT / 16, 256, 0, stream>>>(uacc, u_w, alpha,
                                                      multi_u, single_u, U_CNT);
    gemm_prelu_kernel<<<I_CNT / 16, 256, 0, stream>>>(iacc, i_w, alpha,
                                                      multi_i, single_i, I_CNT);
}